// GATModel_69784628625436
// MI455X (gfx1250) — compile-verified
//
#include <hip/hip_runtime.h>

#define N_NODES 50000
#define N_EDGES 800000

typedef __attribute__((ext_vector_type(2))) float v2f;
typedef __attribute__((ext_vector_type(8))) float v8f;

__device__ __forceinline__ float lrelu(float x) { return x > 0.f ? x : 0.2f * x; }

// monotone float <-> uint mapping for atomicMax over signed floats
__device__ __forceinline__ unsigned ford(float f) {
    unsigned u = __float_as_uint(f);
    return (u & 0x80000000u) ? ~u : (u | 0x80000000u);
}
__device__ __forceinline__ float funord(unsigned u) {
    u = (u & 0x80000000u) ? (u & 0x7fffffffu) : ~u;
    return __uint_as_float(u);
}

// ---------------------------------------------------------------------------
// Fused dual GEMM: outq = H @ Wq, outk = H @ Ws   (rows x 64, K = 64), fp32 WMMA.
// One wave per 16-row tile; computes all 4 column tiles for both outputs,
// reading each A fragment once.
// ---------------------------------------------------------------------------
__global__ void gemm_qk(const float* __restrict__ Hm, const float* __restrict__ Wq,
                        const float* __restrict__ Ws, float* __restrict__ outq,
                        float* __restrict__ outk, int nrows) {
    int lane = threadIdx.x & 31;
    int wid  = (blockIdx.x * blockDim.x + threadIdx.x) >> 5;
    int r0 = wid * 16;
    if (r0 >= nrows) return;
    int m    = lane & 15;          // row within tile (A), col within tile (B/D)
    int koff = (lane >> 4) << 1;   // K sub-offset per half-wave
    v8f accq[4] = {};
    v8f acck[4] = {};
    const float* arow = Hm + (size_t)(r0 + m) * 64;
    for (int k = 0; k < 64; k += 4) {
        v2f a = *(const v2f*)(arow + k + koff);
#pragma unroll
        for (int ct = 0; ct < 4; ++ct) {
            int n = ct * 16 + m;
            v2f bq, bs;
            bq.x = Wq[(k + koff) * 64 + n];
            bq.y = Wq[(k + koff + 1) * 64 + n];
            bs.x = Ws[(k + koff) * 64 + n];
            bs.y = Ws[(k + koff + 1) * 64 + n];
            accq[ct] = __builtin_amdgcn_wmma_f32_16x16x4_f32(false, a, false, bq,
                                                             (short)0, accq[ct], false, false);
            acck[ct] = __builtin_amdgcn_wmma_f32_16x16x4_f32(false, a, false, bs,
                                                             (short)0, acck[ct], false, false);
        }
    }
    int rbase = r0 + ((lane >> 4) << 3);   // D: vgpr i -> row i (+8 for hi half-wave)
#pragma unroll
    for (int ct = 0; ct < 4; ++ct)
#pragma unroll
        for (int i = 0; i < 8; ++i) {
            outq[(size_t)(rbase + i) * 64 + ct * 16 + m] = accq[ct][i];
            outk[(size_t)(rbase + i) * 64 + ct * 16 + m] = acck[ct][i];
        }
}

// ---------------------------------------------------------------------------
// Dense layer: out = relu( concat[A0, relu(A1)] @ W + bias ),  K=128, N=64.
// ---------------------------------------------------------------------------
__global__ void dense_relu(const float* __restrict__ A0, const float* __restrict__ A1,
                           const float* __restrict__ W, const float* __restrict__ bias,
                           float* __restrict__ out, int nrows) {
    int lane = threadIdx.x & 31;
    int wid  = (blockIdx.x * blockDim.x + threadIdx.x) >> 5;
    int r0 = wid * 16;
    if (r0 >= nrows) return;
    int m    = lane & 15;
    int koff = (lane >> 4) << 1;
    v8f acc[4] = {};
    const float* a0row = A0 + (size_t)(r0 + m) * 64;
    const float* a1row = A1 + (size_t)(r0 + m) * 64;
    for (int k = 0; k < 128; k += 4) {
        v2f a;
        if (k < 64) {
            a = *(const v2f*)(a0row + k + koff);
        } else {
            v2f t = *(const v2f*)(a1row + (k - 64) + koff);
            a.x = fmaxf(t.x, 0.f);            // relu on pooled messages
            a.y = fmaxf(t.y, 0.f);
        }
#pragma unroll
        for (int ct = 0; ct < 4; ++ct) {
            int n = ct * 16 + m;
            v2f b;
            b.x = W[(k + koff) * 64 + n];
            b.y = W[(k + koff + 1) * 64 + n];
            acc[ct] = __builtin_amdgcn_wmma_f32_16x16x4_f32(false, a, false, b,
                                                            (short)0, acc[ct], false, false);
        }
    }
    int rbase = r0 + ((lane >> 4) << 3);
#pragma unroll
    for (int ct = 0; ct < 4; ++ct) {
        float bv = bias[ct * 16 + m];
#pragma unroll
        for (int i = 0; i < 8; ++i) {
            float v = acc[ct][i] + bv;
            out[(size_t)(rbase + i) * 64 + ct * 16 + m] = fmaxf(v, 0.f);
        }
    }
}

// ---------------------------------------------------------------------------
// Zero scratch accumulators.
// ---------------------------------------------------------------------------
__global__ void zero_bufs(float* pooled, float* segsum, unsigned* segmax,
                          int npool, int nseg) {
    int stride = gridDim.x * blockDim.x;
    for (int i = blockIdx.x * blockDim.x + threadIdx.x; i < npool; i += stride)
        pooled[i] = 0.f;
    for (int i = blockIdx.x * blockDim.x + threadIdx.x; i < nseg; i += stride) {
        segsum[i] = 0.f;
        segmax[i] = 0u;   // ford^-1(0) is below ford(any finite float)
    }
}

// ---------------------------------------------------------------------------
// Pass A: per-edge attention logits + segment max.
// One wave per edge; lane owns channels {2*lane, 2*lane+1}. k recomputed as
// hk[sender] + We^T e  (We staged in LDS, edge feats exchanged via shfl).
// ---------------------------------------------------------------------------
template <int HEADS, int CH>
__global__ void edge_attn_logits(const float* __restrict__ q, const float* __restrict__ khs,
                                 const float* __restrict__ ef, const float* __restrict__ We,
                                 const float* __restrict__ av, const int* __restrict__ snd,
                                 const int* __restrict__ rcv, float* __restrict__ logits,
                                 unsigned* __restrict__ segmax, int E) {
    __shared__ float sWe[32 * 64];
    __shared__ float sA[64];
    for (int i = threadIdx.x; i < 32 * 64; i += blockDim.x) sWe[i] = We[i];
    for (int i = threadIdx.x; i < 64; i += blockDim.x) sA[i] = av[i];
    __syncthreads();
    int lane = threadIdx.x & 31;
    int wid  = (blockIdx.x * blockDim.x + threadIdx.x) >> 5;
    int nw   = (gridDim.x * blockDim.x) >> 5;
    int c0 = lane * 2;
    for (int e = wid; e < E; e += nw) {
        int s = snd[e];
        int r = rcv[e];
        float ve = ef[e * 32 + lane];
        v2f kk = *(const v2f*)(khs + (size_t)s * 64 + c0);
        float k0 = kk.x, k1 = kk.y;
#pragma unroll
        for (int d = 0; d < 32; ++d) {
            float ed = __shfl(ve, d, 32);
            k0 = fmaf(ed, sWe[d * 64 + c0], k0);
            k1 = fmaf(ed, sWe[d * 64 + c0 + 1], k1);
        }
        v2f qq = *(const v2f*)(q + (size_t)r * 64 + c0);
        float f0 = lrelu(qq.x + k0);
        float f1 = lrelu(qq.y + k1);
        float p = f0 * sA[c0] + f1 * sA[c0 + 1];
#pragma unroll
        for (int off = 1; off < CH / 2; off <<= 1) p += __shfl_xor(p, off, 32);
        if ((lane & (CH / 2 - 1)) == 0) {
            int h = lane / (CH / 2);
            logits[(size_t)e * HEADS + h] = p;
            atomicMax(segmax + r * HEADS + h, ford(p));
        }
    }
}

// ---------------------------------------------------------------------------
// Pass B: exp(logit - segmax) + segment sum.
// ---------------------------------------------------------------------------
template <int HEADS>
__global__ void edge_softmax_num(const int* __restrict__ rcv, float* __restrict__ logits,
                                 const unsigned* __restrict__ segmax,
                                 float* __restrict__ segsum, int E) {
    int stride = gridDim.x * blockDim.x;
    int total = E * HEADS;
    for (int idx = blockIdx.x * blockDim.x + threadIdx.x; idx < total; idx += stride) {
        int e = idx / HEADS;
        int h = idx - e * HEADS;
        int r = rcv[e];
        float m = funord(segmax[r * HEADS + h]);
        float ex = expf(logits[idx] - m);
        logits[idx] = ex;
        atomicAdd(segsum + r * HEADS + h, ex);
    }
}

// ---------------------------------------------------------------------------
// Pass C: msg = k * alpha, scatter-add to receiver (L2-resident pooled buffer).
// ---------------------------------------------------------------------------
template <int HEADS, int CH>
__global__ void edge_aggregate(const float* __restrict__ khs, const float* __restrict__ ef,
                               const float* __restrict__ We, const int* __restrict__ snd,
                               const int* __restrict__ rcv, const float* __restrict__ logits,
                               const float* __restrict__ segsum, float* __restrict__ pooled,
                               int E) {
    __shared__ float sWe[32 * 64];
    for (int i = threadIdx.x; i < 32 * 64; i += blockDim.x) sWe[i] = We[i];
    __syncthreads();
    int lane = threadIdx.x & 31;
    int wid  = (blockIdx.x * blockDim.x + threadIdx.x) >> 5;
    int nw   = (gridDim.x * blockDim.x) >> 5;
    int c0 = lane * 2;
    int h  = c0 / CH;
    for (int e = wid; e < E; e += nw) {
        int s = snd[e];
        int r = rcv[e];
        float ve = ef[e * 32 + lane];
        v2f kk = *(const v2f*)(khs + (size_t)s * 64 + c0);
        float k0 = kk.x, k1 = kk.y;
#pragma unroll
        for (int d = 0; d < 32; ++d) {
            float ed = __shfl(ve, d, 32);
            k0 = fmaf(ed, sWe[d * 64 + c0], k0);
            k1 = fmaf(ed, sWe[d * 64 + c0 + 1], k1);
        }
        float alpha = logits[(size_t)e * HEADS + h] / segsum[r * HEADS + h];
        atomicAdd(pooled + (size_t)r * 64 + c0, k0 * alpha);
        atomicAdd(pooled + (size_t)r * 64 + c0 + 1, k1 * alpha);
    }
}

// ---------------------------------------------------------------------------
// Readout: out[row] = h2[row,:] . Wd + bd   (wave per row, shfl reduction)
// ---------------------------------------------------------------------------
__global__ void readout(const float* __restrict__ h, const float* __restrict__ Wd,
                        const float* __restrict__ bd, float* __restrict__ out, int n) {
    int lane = threadIdx.x & 31;
    int wid  = (blockIdx.x * blockDim.x + threadIdx.x) >> 5;
    int nw   = (gridDim.x * blockDim.x) >> 5;
    v2f w = *(const v2f*)(Wd + lane * 2);
    for (int row = wid; row < n; row += nw) {
        v2f v = *(const v2f*)(h + (size_t)row * 64 + lane * 2);
        float p = v.x * w.x + v.y * w.y;
#pragma unroll
        for (int off = 16; off > 0; off >>= 1) p += __shfl_xor(p, off, 32);
        if (lane == 0) out[row] = p + bd[0];
    }
}

extern "C" void kernel_launch(void* const* d_in, const int* in_sizes, int n_in,
                              void* d_out, int out_size, void* d_ws, size_t ws_size,
                              hipStream_t stream) {
    const float* node_feats = (const float*)d_in[0];
    const float* edge_feats = (const float*)d_in[1];
    const int*   senders    = (const int*)d_in[2];
    const int*   receivers  = (const int*)d_in[3];
    const float* Wq1 = (const float*)d_in[4];
    const float* Ws1 = (const float*)d_in[5];
    const float* We1 = (const float*)d_in[6];
    const float* a1  = (const float*)d_in[7];
    const float* Wn1 = (const float*)d_in[8];
    const float* bn1 = (const float*)d_in[9];
    const float* Wq2 = (const float*)d_in[10];
    const float* Ws2 = (const float*)d_in[11];
    const float* We2 = (const float*)d_in[12];
    const float* a2  = (const float*)d_in[13];
    const float* Wn2 = (const float*)d_in[14];
    const float* bn2 = (const float*)d_in[15];
    const float* Wd  = (const float*)d_in[16];
    const float* bd  = (const float*)d_in[17];
    float* outp = (float*)d_out;

    const int N = N_NODES, E = N_EDGES;

    // workspace layout (floats)
    float* hq     = (float*)d_ws;                 // N*64
    float* hk     = hq + (size_t)N * 64;          // N*64
    float* pooled = hk + (size_t)N * 64;          // N*64
    float* h1     = pooled + (size_t)N * 64;      // N*64
    float* h2     = h1 + (size_t)N * 64;          // N*64
    float* logits = h2 + (size_t)N * 64;          // E*4
    float* segsum = logits + (size_t)E * 4;       // N*4
    unsigned* segmax = (unsigned*)(segsum + (size_t)N * 4);  // N*4

    const int ntiles = (N + 15) / 16;                  // 3125 row tiles
    dim3 gblk(128);                                    // 4 waves/block
    dim3 ggrd((ntiles + 3) / 4);
    dim3 eblk(256);                                    // 8 waves/block
    dim3 egrd(4096);
    dim3 zgrd(2048);

    // ---- layer 1 ----
    gemm_qk<<<ggrd, gblk, 0, stream>>>(node_feats, Wq1, Ws1, hq, hk, N);
    zero_bufs<<<zgrd, eblk, 0, stream>>>(pooled, segsum, segmax, N * 64, N * 4);
    edge_attn_logits<4, 16><<<egrd, eblk, 0, stream>>>(hq, hk, edge_feats, We1, a1,
                                                       senders, receivers, logits, segmax, E);
    edge_softmax_num<4><<<egrd, eblk, 0, stream>>>(receivers, logits, segmax, segsum, E);
    edge_aggregate<4, 16><<<egrd, eblk, 0, stream>>>(hk, edge_feats, We1, senders, receivers,
                                                     logits, segsum, pooled, E);
    dense_relu<<<ggrd, gblk, 0, stream>>>(node_feats, pooled, Wn1, bn1, h1, N);

    // ---- layer 2 ----
    gemm_qk<<<ggrd, gblk, 0, stream>>>(h1, Wq2, Ws2, hq, hk, N);
    zero_bufs<<<zgrd, eblk, 0, stream>>>(pooled, segsum, segmax, N * 64, N * 4);
    edge_attn_logits<1, 64><<<egrd, eblk, 0, stream>>>(hq, hk, edge_feats, We2, a2,
                                                       senders, receivers, logits, segmax, E);
    edge_softmax_num<1><<<egrd, eblk, 0, stream>>>(receivers, logits, segmax, segsum, E);
    edge_aggregate<1, 64><<<egrd, eblk, 0, stream>>>(hk, edge_feats, We2, senders, receivers,
                                                     logits, segsum, pooled, E);
    dense_relu<<<ggrd, gblk, 0, stream>>>(h1, pooled, Wn2, bn2, h2, N);

    // ---- readout ----
    readout<<<egrd, eblk, 0, stream>>>(h2, Wd, bd, outp, N);
    (void)in_sizes; (void)n_in; (void)out_size; (void)ws_size;
}